// LocalConvTrans2d_74526272520628
// MI455X (gfx1250) — compile-verified
//
#include <hip/hip_runtime.h>

typedef __attribute__((ext_vector_type(2))) float v2f;
typedef __attribute__((ext_vector_type(8))) float v8f;

#define BS    64    // batch (GEMM M)
#define IN_C  256   // channels (GEMM K)
#define HW    784   // 28*28 locations
#define W28   28
#define OCKK  1024  // 64 out_c * 4 * 4 (GEMM N)
#define OUTW  31    // 28 + 4 - 1

// One workgroup per spatial location l.
// 8 waves; each wave owns N-tiles {wave, wave+8, ...} and computes all 4
// M-tiles per N-tile so each B fragment is loaded from HBM exactly once.
__global__ __launch_bounds__(256, 2)
void lc_gemm_fold_kernel(const float* __restrict__ x,
                         const float* __restrict__ w,
                         float* __restrict__ out) {
    const int l    = blockIdx.x;          // location
    const int oi   = l / W28;             // fold row base
    const int oj   = l % W28;             // fold col base
    const int tid  = threadIdx.x;
    const int wave = tid >> 5;
    const int lane = tid & 31;
    const int hi   = lane >> 4;           // half-wave: 0 or 1
    const int lo   = lane & 15;

    __shared__ float lds_a[BS * IN_C];    // 64 KB: x[:, :, l] as [b][c]

    // ---- Stage A into LDS (x fits in 192MB L2, so these re-reads are cheap)
    const float* xl = x + l;
    #pragma unroll
    for (int t = 0; t < (BS * IN_C) / 256; ++t) {
        int idx = tid + t * 256;                 // idx = b*IN_C + c
        lds_a[idx] = xl[(size_t)idx * HW];
    }
    __syncthreads();

    const float* wl = w + (size_t)l * IN_C * OCKK;

    for (int nt = wave; nt < OCKK / 16; nt += 8) {
        const int ncol = nt * 16 + lo;           // this lane's B/C column

        v8f acc[4];
        acc[0] = (v8f){}; acc[1] = (v8f){}; acc[2] = (v8f){}; acc[3] = (v8f){};

        #pragma unroll 2
        for (int k0 = 0; k0 < IN_C; k0 += 4) {
            const int krow = k0 + 2 * hi;        // this lane's K rows (krow, krow+1)

            // B fragment: 4x16 f32, streamed once from HBM (non-temporal:
            // don't let the 822MB weight stream evict x from L2)
            v2f bf;
            bf.x = __builtin_nontemporal_load(&wl[(size_t)krow       * OCKK + ncol]);
            bf.y = __builtin_nontemporal_load(&wl[(size_t)(krow + 1) * OCKK + ncol]);

            // A fragments from LDS: lane's two K values are contiguous -> b64,
            // one row per lane (m = lo), rows IN_C floats apart
            v2f a0 = *(const v2f*)&lds_a[(0 * 16 + lo) * IN_C + krow];
            v2f a1 = *(const v2f*)&lds_a[(1 * 16 + lo) * IN_C + krow];
            v2f a2 = *(const v2f*)&lds_a[(2 * 16 + lo) * IN_C + krow];
            v2f a3 = *(const v2f*)&lds_a[(3 * 16 + lo) * IN_C + krow];

            acc[0] = __builtin_amdgcn_wmma_f32_16x16x4_f32(false, a0, false, bf,
                                                           (short)0, acc[0], false, false);
            acc[1] = __builtin_amdgcn_wmma_f32_16x16x4_f32(false, a1, false, bf,
                                                           (short)0, acc[1], false, false);
            acc[2] = __builtin_amdgcn_wmma_f32_16x16x4_f32(false, a2, false, bf,
                                                           (short)0, acc[2], false, false);
            acc[3] = __builtin_amdgcn_wmma_f32_16x16x4_f32(false, a3, false, bf,
                                                           (short)0, acc[3], false, false);
        }

        // ---- Fold (overlap-add) scatter. d = oc*16 + kh*4 + kw.
        const int d  = ncol;
        const int oc = d >> 4;
        const int kh = (d >> 2) & 3;
        const int kw = d & 3;
        const int oy = oi + kh;
        const int ox = oj + kw;

        #pragma unroll
        for (int mt = 0; mt < 4; ++mt) {
            #pragma unroll
            for (int v = 0; v < 8; ++v) {
                const int b = mt * 16 + hi * 8 + v;   // C layout: m = 8*(lane/16)+v
                const size_t idx = (((size_t)b * 64 + oc) * OUTW + oy) * OUTW + ox;
                atomicAdd(&out[idx], acc[mt][v]);
            }
        }
    }
}

extern "C" void kernel_launch(void* const* d_in, const int* in_sizes, int n_in,
                              void* d_out, int out_size, void* d_ws, size_t ws_size,
                              hipStream_t stream) {
    const float* x = (const float*)d_in[0];   // [64, 256, 28, 28]
    const float* w = (const float*)d_in[1];   // [784, 256, 1024]
    float* out = (float*)d_out;               // [64, 64, 31, 31]

    // Zero output (harness poisons it); graph-capture-safe async memset.
    hipMemsetAsync(out, 0, (size_t)out_size * sizeof(float), stream);

    lc_gemm_fold_kernel<<<dim3(HW), dim3(256), 0, stream>>>(x, w, out);
}